// PointNetSetAbstraction_7825430413396
// MI455X (gfx1250) — compile-verified
//
#include <hip/hip_runtime.h>
#include <hip/hip_bf16.h>

#define NPTS   16384
#define NPOINT 1024
#define NSAMP  32
#define NB     4
#define DPTS   64
#define K1RAW  67
#define K1PAD  68
#define MTOT   (NB * NPOINT * NSAMP)   // 131072
#define R2     0.09f
#define EPSBN  1e-5f

typedef float v2f __attribute__((ext_vector_type(2)));
typedef float v8f __attribute__((ext_vector_type(8)));

// ---------------------------------------------------------------------------
// 0) zero the BN sum accumulators (512 floats)
// ---------------------------------------------------------------------------
__global__ void init_sums_kernel(float* __restrict__ sums) {
    int t = blockIdx.x * blockDim.x + threadIdx.x;
    if (t < 512) sums[t] = 0.0f;
}

// ---------------------------------------------------------------------------
// 1) Farthest point sampling: 1 block per batch, 1024 threads, 16 pts/thread
//    held in registers. Argmax via wave shuffle + 1 LDS hop (2 barriers/iter).
// ---------------------------------------------------------------------------
__global__ __launch_bounds__(1024) void fps_kernel(const float* __restrict__ xyz,
                                                   int* __restrict__ fps) {
    const int b = blockIdx.x;
    const int t = threadIdx.x;
    const float* px = xyz + (size_t)b * 3 * NPTS;

    float lx[16], ly[16], lz[16], ld[16];
#pragma unroll
    for (int j = 0; j < 16; ++j) {
        int n = j * 1024 + t;
        lx[j] = px[n];
        ly[j] = px[NPTS + n];
        lz[j] = px[2 * NPTS + n];
        ld[j] = 1e10f;
    }

    __shared__ float wv[32];
    __shared__ int   wi[32];
    __shared__ int   scur;
    __shared__ float sc[3];

    const int wave = t >> 5, lane = t & 31;
    int cur = 0;

    for (int it = 0; it < NPOINT; ++it) {
        if (t == 0) fps[(b << 10) + it] = cur;
        if (t < 3)  sc[t] = px[t * NPTS + cur];
        __syncthreads();
        const float cx = sc[0], cy = sc[1], cz = sc[2];

        float bv = -1.0f;
        int   bi = 0x7fffffff;
#pragma unroll
        for (int j = 0; j < 16; ++j) {
            float dx = lx[j] - cx, dy = ly[j] - cy, dz = lz[j] - cz;
            float d  = dx * dx + dy * dy + dz * dz;
            float nd = fminf(ld[j], d);
            ld[j] = nd;
            int n = j * 1024 + t;
            if (nd > bv || (nd == bv && n < bi)) { bv = nd; bi = n; }
        }
        // wave32 argmax reduce (first-index tie break)
#pragma unroll
        for (int off = 16; off; off >>= 1) {
            float ov = __shfl_xor(bv, off, 32);
            int   oi = __shfl_xor(bi, off, 32);
            if (ov > bv || (ov == bv && oi < bi)) { bv = ov; bi = oi; }
        }
        if (lane == 0) { wv[wave] = bv; wi[wave] = bi; }
        __syncthreads();
        if (wave == 0) {
            bv = wv[lane]; bi = wi[lane];
#pragma unroll
            for (int off = 16; off; off >>= 1) {
                float ov = __shfl_xor(bv, off, 32);
                int   oi = __shfl_xor(bi, off, 32);
                if (ov > bv || (ov == bv && oi < bi)) { bv = ov; bi = oi; }
            }
            if (lane == 0) scur = bi;
        }
        __syncthreads();
        cur = scur;
    }
}

// ---------------------------------------------------------------------------
// 2) Gather new_xyz: write output slab (B,3,NPOINT) + (B*NPOINT,3) scratch
// ---------------------------------------------------------------------------
__global__ void gather_newxyz_kernel(const float* __restrict__ xyz,
                                     const int* __restrict__ fps,
                                     float* __restrict__ out,   // (B,3,NPOINT)
                                     float* __restrict__ nxw) { // (B*NPOINT,3)
    int t = blockIdx.x * blockDim.x + threadIdx.x;   // b*1024 + s
    if (t >= NB * NPOINT) return;
    int b = t >> 10, s = t & 1023;
    int p = fps[t];
#pragma unroll
    for (int c = 0; c < 3; ++c) {
        float v = xyz[((size_t)(b * 3 + c) << 14) + p];
        out[((b * 3 + c) << 10) + s] = v;
        nxw[t * 3 + c] = v;
    }
}

// ---------------------------------------------------------------------------
// 3) Ball query: one wave per centroid. First NSAMP ascending indices with
//    d^2 <= R2, pad trailing slots with the first hit.
// ---------------------------------------------------------------------------
__global__ __launch_bounds__(256) void ball_query_kernel(const float* __restrict__ xyz,
                                                         const float* __restrict__ nxw,
                                                         int* __restrict__ idx) {
    int wid  = (blockIdx.x * blockDim.x + threadIdx.x) >> 5;  // centroid id
    int lane = threadIdx.x & 31;
    if (wid >= NB * NPOINT) return;
    int b = wid >> 10;
    const float cx = nxw[wid * 3 + 0];
    const float cy = nxw[wid * 3 + 1];
    const float cz = nxw[wid * 3 + 2];
    const float* px = xyz + (size_t)b * 3 * NPTS;
    int* my = idx + (size_t)wid * NSAMP;

    int cnt = 0, first = 0;
    bool haveFirst = false;
    for (int base = 0; base < NPTS && cnt < NSAMP; base += 32) {
        int n = base + lane;
        float dx = px[n] - cx;
        float dy = px[NPTS + n] - cy;
        float dz = px[2 * NPTS + n] - cz;
        bool pred = (dx * dx + dy * dy + dz * dz) <= R2;
        unsigned long long mask = __ballot(pred);
        if (!haveFirst && mask) { first = base + (__ffsll((long long)mask) - 1); haveFirst = true; }
        if (pred) {
            int pos = cnt + __popcll(mask & ((1ull << lane) - 1ull));
            if (pos < NSAMP) my[pos] = n;
        }
        cnt += __popcll(mask);
    }
    if (cnt < NSAMP && lane >= cnt) my[lane] = first;  // centroid itself => cnt>=1
}

// ---------------------------------------------------------------------------
// 4) Build x0 (M x 68): [rel xyz (3) | grouped points (64) | zero pad (1)]
// ---------------------------------------------------------------------------
__global__ __launch_bounds__(256) void build_x0_kernel(const float* __restrict__ xyz,
                                                       const float* __restrict__ pts,
                                                       const int* __restrict__ idx,
                                                       const float* __restrict__ nxw,
                                                       float* __restrict__ x0) {
    int g = blockIdx.x;             // b*1024 + s
    int b = g >> 10;
    __shared__ int   sidx[NSAMP];
    __shared__ float ctr[3];
    if (threadIdx.x < NSAMP) sidx[threadIdx.x] = idx[(size_t)g * NSAMP + threadIdx.x];
    if (threadIdx.x < 3)     ctr[threadIdx.x]  = nxw[g * 3 + threadIdx.x];
    __syncthreads();
    float* xr = x0 + (size_t)g * NSAMP * K1PAD;
    for (int e = threadIdx.x; e < NSAMP * K1PAD; e += 256) {
        int k = e / K1PAD, c = e - k * K1PAD;
        int p = sidx[k];
        float v;
        if (c < 3)          v = xyz[((size_t)(b * 3 + c) << 14) + p] - ctr[c];
        else if (c < K1RAW) v = pts[((size_t)(b * DPTS + (c - 3)) << 14) + p];
        else                v = 0.0f;
        xr[e] = v;
    }
}

// ---------------------------------------------------------------------------
// 5) GEMM via V_WMMA_F32_16X16X4_F32, fully specialized at compile time.
//    Each wave owns one 16-col tile x 4 row-tiles (64 rows). B fragments for
//    the whole K range are preloaded from LDS into registers (NK x v2f) and
//    reused across the 4 row tiles; the 4 accumulator chains are independent
//    so the scheduler can hide the A global-load latency between v_wmma ops.
//    Optional batchnorm+ReLU is fused into the A-fragment load (DOBN).
//    A layout:  lane l holds rows M=row+(l&15), cols K = 4k+2*(l>>4)+{0,1}
//    B layout:  lane l holds col N=ct*16+(l&15), rows K = 4k+2*(l>>4)+{0,1}
//    C layout:  acc[r] at lane l is element (row + r + 8*(l>>4), ct*16+(l&15))
// ---------------------------------------------------------------------------
template <int NK, int KRAW, int NOUT, bool DOBN>
__global__ __launch_bounds__(256) void gemm_wmma_kernel(
    const float* __restrict__ A,      // M x (NK*4) row-major
    const float* __restrict__ W,      // NOUT x KRAW row-major
    const float* __restrict__ bias,   // NOUT
    float* __restrict__ Y,            // M x NOUT
    const float* __restrict__ bnsum,  // KRAW (input-channel BN stats)
    const float* __restrict__ bnsumsq,
    const float* __restrict__ bng,
    const float* __restrict__ bnb)
{
    constexpr int KDIM = NK * 4;
    __shared__ float wT[KDIM * NOUT];     // transposed weights, zero-padded K
    __shared__ float sscale[KDIM];
    __shared__ float sshift[KDIM];

#pragma unroll 4
    for (int e = threadIdx.x; e < KDIM * NOUT; e += 256) {
        int c = e / NOUT, o = e - c * NOUT;
        wT[e] = (c < KRAW) ? W[o * KRAW + c] : 0.0f;
    }
    if (DOBN) {
        const float invM = 1.0f / (float)MTOT;
        for (int c = threadIdx.x; c < KRAW; c += 256) {
            float mean = bnsum[c] * invM;
            float var  = bnsumsq[c] * invM - mean * mean;
            float scv  = bng[c] * rsqrtf(var + EPSBN);
            sscale[c]  = scv;
            sshift[c]  = bnb[c] - mean * scv;
        }
    }
    __syncthreads();

    const int wave = threadIdx.x >> 5, lane = threadIdx.x & 31;
    constexpr int colTiles = NOUT / 16;
    constexpr int rsPerBlock = 8 / colTiles;            // row-supertiles/block
    const int ct = wave % colTiles;
    const int rs = wave / colTiles;
    const int row0 = (blockIdx.x * rsPerBlock + rs) * 64;   // 4 row tiles
    const int half = lane >> 4, l16 = lane & 15;
    const int o = (ct << 4) + l16;

    // preload all B fragments into registers (immediate ds_load offsets)
    v2f bReg[NK];
#pragma unroll
    for (int kk = 0; kk < NK; ++kk) {
        int c0 = kk * 4 + 2 * half;
        bReg[kk] = {wT[c0 * NOUT + o], wT[(c0 + 1) * NOUT + o]};
    }
    const float bv = bias[o];

#pragma unroll
    for (int rt = 0; rt < 4; ++rt) {
        const float* arow = A + (size_t)(row0 + rt * 16 + l16) * KDIM + 2 * half;
        if (rt < 3)
            __builtin_prefetch(arow + 16 * KDIM, 0, 0);   // global_prefetch_b8
        v8f acc;
#pragma unroll
        for (int r = 0; r < 8; ++r) acc[r] = bv;
#pragma unroll
        for (int kk = 0; kk < NK; ++kk) {
            float a0 = arow[kk * 4];
            float a1 = arow[kk * 4 + 1];
            if (DOBN) {
                int c0 = kk * 4 + 2 * half;
                a0 = fmaxf(a0 * sscale[c0]     + sshift[c0],     0.0f);
                a1 = fmaxf(a1 * sscale[c0 + 1] + sshift[c0 + 1], 0.0f);
            }
            v2f a = {a0, a1};
            acc = __builtin_amdgcn_wmma_f32_16x16x4_f32(false, a, false, bReg[kk],
                                                        (short)0, acc, false, false);
        }
#pragma unroll
        for (int r = 0; r < 8; ++r) {
            int m = row0 + rt * 16 + r + (half << 3);
            Y[(size_t)m * NOUT + o] = acc[r];
        }
    }
}

// ---------------------------------------------------------------------------
// 6) Per-channel sum / sumsq reduction (for batchnorm statistics)
// ---------------------------------------------------------------------------
template <int C>
__global__ __launch_bounds__(256) void reduce_bn_kernel(const float* __restrict__ Y,
                                                        float* __restrict__ sum,
                                                        float* __restrict__ sumsq,
                                                        int rowsPerBlock) {
    const int c  = threadIdx.x % C;
    const int rl = threadIdx.x / C;
    constexpr int rpt = 256 / C;
    const int row0 = blockIdx.x * rowsPerBlock;
    float s = 0.0f, ss = 0.0f;
    for (int r = rl; r < rowsPerBlock; r += rpt) {
        float v = Y[(size_t)(row0 + r) * C + c];
        s += v; ss += v * v;
    }
    __shared__ float ls[256];
    __shared__ float lss[256];
    ls[threadIdx.x] = s; lss[threadIdx.x] = ss;
    __syncthreads();
#pragma unroll
    for (int off = 128; off >= C; off >>= 1) {
        if (threadIdx.x < (unsigned)off) {
            ls[threadIdx.x]  += ls[threadIdx.x + off];
            lss[threadIdx.x] += lss[threadIdx.x + off];
        }
        __syncthreads();
    }
    if ((int)threadIdx.x < C) {
        atomicAdd(&sum[c],   ls[threadIdx.x]);
        atomicAdd(&sumsq[c], lss[threadIdx.x]);
    }
}

// ---------------------------------------------------------------------------
// 7) Final: BN3 + ReLU + max over the 32 samples, transpose to (B,128,NPOINT)
// ---------------------------------------------------------------------------
__global__ __launch_bounds__(256) void final_max_kernel(const float* __restrict__ y3,
                                                        const float* __restrict__ sum,
                                                        const float* __restrict__ sumsq,
                                                        const float* __restrict__ g,
                                                        const float* __restrict__ beta,
                                                        float* __restrict__ out) {
    int t = blockIdx.x * blockDim.x + threadIdx.x;  // bs*128 + o
    if (t >= NB * NPOINT * 128) return;
    int o  = t & 127;
    int bs = t >> 7;
    const float invM = 1.0f / (float)MTOT;
    float mean = sum[o] * invM;
    float var  = sumsq[o] * invM - mean * mean;
    float sc   = g[o] * rsqrtf(var + EPSBN);
    float sh   = beta[o] - mean * sc;
    float m = -1e30f;
#pragma unroll 4
    for (int k = 0; k < NSAMP; ++k) {
        float v = y3[((size_t)bs * NSAMP + k) * 128 + o];
        m = fmaxf(m, fmaxf(v * sc + sh, 0.0f));
    }
    int b = bs >> 10, s = bs & 1023;
    out[(size_t)NB * 3 * NPOINT + ((size_t)(b * 128 + o) << 10) + s] = m;
}

// ---------------------------------------------------------------------------
// launch
// ---------------------------------------------------------------------------
extern "C" void kernel_launch(void* const* d_in, const int* in_sizes, int n_in,
                              void* d_out, int out_size, void* d_ws, size_t ws_size,
                              hipStream_t stream) {
    (void)in_sizes; (void)n_in; (void)out_size; (void)ws_size;
    const float* xyz = (const float*)d_in[0];   // (B,3,N)
    const float* pts = (const float*)d_in[1];   // (B,D,N)
    const float* w1  = (const float*)d_in[2];
    const float* b1  = (const float*)d_in[3];
    const float* g1  = (const float*)d_in[4];
    const float* bt1 = (const float*)d_in[5];
    const float* w2  = (const float*)d_in[6];
    const float* b2  = (const float*)d_in[7];
    const float* g2  = (const float*)d_in[8];
    const float* bt2 = (const float*)d_in[9];
    const float* w3  = (const float*)d_in[10];
    const float* b3  = (const float*)d_in[11];
    const float* g3  = (const float*)d_in[12];
    const float* bt3 = (const float*)d_in[13];
    float* out = (float*)d_out;

    char* ws = (char*)d_ws;
    size_t off = 0;
    auto alloc = [&](size_t bytes) { void* p = ws + off; off = (off + bytes + 255) & ~(size_t)255; return p; };
    int*   fps   = (int*)  alloc((size_t)NB * NPOINT * 4);
    int*   idx   = (int*)  alloc((size_t)NB * NPOINT * NSAMP * 4);
    float* nxw   = (float*)alloc((size_t)NB * NPOINT * 3 * 4);
    float* sums  = (float*)alloc(512 * 4);
    float* x0    = (float*)alloc((size_t)MTOT * K1PAD * 4);   // reused as y2
    float* y1    = (float*)alloc((size_t)MTOT * 64 * 4);
    float* y3    = (float*)alloc((size_t)MTOT * 128 * 4);
    float* y2    = x0;   // x0 dead after GEMM1

    float* sum1 = sums +   0; float* ss1 = sums +  64;
    float* sum2 = sums + 128; float* ss2 = sums + 192;
    float* sum3 = sums + 256; float* ss3 = sums + 384;

    init_sums_kernel<<<2, 256, 0, stream>>>(sums);
    fps_kernel<<<NB, 1024, 0, stream>>>(xyz, fps);
    gather_newxyz_kernel<<<(NB * NPOINT) / 256, 256, 0, stream>>>(xyz, fps, out, nxw);
    ball_query_kernel<<<(NB * NPOINT * 32) / 256, 256, 0, stream>>>(xyz, nxw, idx);
    build_x0_kernel<<<NB * NPOINT, 256, 0, stream>>>(xyz, pts, idx, nxw, x0);

    // layer 1: x0 (M x 68) * w1^T -> y1 (M x 64)     [block: 128 rows]
    gemm_wmma_kernel<17, K1RAW, 64, false><<<MTOT / 128, 256, 0, stream>>>(
        x0, w1, b1, y1, nullptr, nullptr, nullptr, nullptr);
    reduce_bn_kernel<64><<<MTOT / 1024, 256, 0, stream>>>(y1, sum1, ss1, 1024);

    // layer 2: BN1+ReLU(y1) * w2^T -> y2 (M x 64)    [block: 128 rows]
    gemm_wmma_kernel<16, 64, 64, true><<<MTOT / 128, 256, 0, stream>>>(
        y1, w2, b2, y2, sum1, ss1, g1, bt1);
    reduce_bn_kernel<64><<<MTOT / 1024, 256, 0, stream>>>(y2, sum2, ss2, 1024);

    // layer 3: BN2+ReLU(y2) * w3^T -> y3 (M x 128)   [block: 64 rows]
    gemm_wmma_kernel<16, 64, 128, true><<<MTOT / 64, 256, 0, stream>>>(
        y2, w3, b3, y3, sum2, ss2, g2, bt2);
    reduce_bn_kernel<128><<<MTOT / 1024, 256, 0, stream>>>(y3, sum3, ss3, 1024);

    final_max_kernel<<<(NB * NPOINT * 128) / 256, 256, 0, stream>>>(y3, sum3, ss3, g3, bt3, out);
}